// HetDistMultDecoder_35983236006033
// MI455X (gfx1250) — compile-verified
//
#include <hip/hip_runtime.h>

typedef __attribute__((ext_vector_type(2))) float v2f;
typedef __attribute__((ext_vector_type(8))) float v8f;

constexpr int kH      = 128;   // hidden dim
constexpr int kR      = 17;    // num relations + 1
constexpr int kRPad   = 32;    // padded N (two 16-wide WMMA tiles)
constexpr int kStride = 132;   // LDS row stride: 128 + 4-float skew (bank-conflict free)
constexpr int kTileM  = 16;    // edges per wave-tile
constexpr int kWaves  = 4;     // waves per block (wave32)

__global__ __launch_bounds__(128) void hetdistmult_wmma(
    const float* __restrict__ z,          // [NUM_NODES][128]
    const float* __restrict__ rel,        // [128][17] row-major
    const long long* __restrict__ posI,   // [2][E]
    const long long* __restrict__ negI,   // [2][E]
    float* __restrict__ out,              // [2E][17]
    long long E)
{
  __shared__ float ldsB[kRPad * kStride];            // rel_emb transposed [r][k]
  __shared__ float ldsP[kWaves * kTileM * kStride];  // per-wave product slab

  const int tid  = threadIdx.x;
  const int lane = tid & 31;
  const int wave = tid >> 5;
  const int khi  = lane >> 4;   // 0 -> K+0/1, 1 -> K+2/3 (f32 WMMA frag halves)
  const int row  = lane & 15;   // M for A, N for B/D

  // Stage rel_emb into LDS transposed to [r][k]; zero-pad relations 17..31.
  for (int i = tid; i < kRPad * kH; i += blockDim.x) {
    int r = i >> 7;          // / kH
    int k = i & (kH - 1);
    ldsB[r * kStride + k] = (r < kR) ? rel[k * kR + r] : 0.0f;
  }
  __syncthreads();

  const long long totalRows = 2 * E;
  const long long tiles = (totalRows + kTileM - 1) / kTileM;
  float* myP = ldsP + wave * kTileM * kStride;

  for (long long t = (long long)blockIdx.x * kWaves + wave; t < tiles;
       t += (long long)gridDim.x * kWaves) {
    const long long rowBase = t * kTileM;
    const bool fullTile = (rowBase + kTileM) <= totalRows;
    const bool oneList  = fullTile && (rowBase >= E || (rowBase + kTileM) <= E);

    // ---- Stage elementwise products z[src]*z[dst] for 16 edges into LDS ----
    if (oneList) {
      // Fast path: whole tile from one edge list, no clamps (the common case).
      const long long* idx = (rowBase < E) ? posI : negI;
      const long long off0 = (rowBase < E) ? rowBase : rowBase - E;
      #pragma unroll 4
      for (int e = 0; e < kTileM; ++e) {
        const long long s = idx[off0 + e];       // uniform -> s_load_b64
        const long long d = idx[off0 + e + E];
        const float4 a = *(const float4*)(z + (size_t)s * kH + lane * 4);
        const float4 b = *(const float4*)(z + (size_t)d * kH + lane * 4);
        float4 p;
        p.x = a.x * b.x; p.y = a.y * b.y; p.z = a.z * b.z; p.w = a.w * b.w;
        *(float4*)(myP + e * kStride + lane * 4) = p;  // ds_store_b128
      }
    } else {
      // Generic path: tail tile and/or pos/neg straddle (rare or absent).
      for (int e = 0; e < kTileM; ++e) {
        long long g = rowBase + e;
        if (g >= totalRows) g = totalRows - 1;   // clamp (replicated rows)
        const long long* idx;
        long long off;
        if (g < E) { idx = posI; off = g; }
        else       { idx = negI; off = g - E; }
        const long long s = idx[off];
        const long long d = idx[off + E];
        const float4 a = *(const float4*)(z + (size_t)s * kH + lane * 4);
        const float4 b = *(const float4*)(z + (size_t)d * kH + lane * 4);
        float4 p;
        p.x = a.x * b.x; p.y = a.y * b.y; p.z = a.z * b.z; p.w = a.w * b.w;
        *(float4*)(myP + e * kStride + lane * 4) = p;
      }
    }
    // LDS is in-order per wave; fence keeps the compiler from hoisting frag
    // reads over the cross-lane product stores.
    asm volatile("s_wait_dscnt 0" ::: "memory");

    // ---- D = A(16x128 products) x B(128x32 rel) via f32 WMMA, K-step 4 ----
    v8f c0 = {0.f, 0.f, 0.f, 0.f, 0.f, 0.f, 0.f, 0.f};  // relations 0..15
    v8f c1 = {0.f, 0.f, 0.f, 0.f, 0.f, 0.f, 0.f, 0.f};  // relations 16..31 (16 valid)
    #pragma unroll 8
    for (int k0 = 0; k0 < kH; k0 += 4) {
      const int ko = k0 + 2 * khi;
      v2f af = *(const v2f*)(myP  +  row       * kStride + ko);  // A frag
      v2f b0 = *(const v2f*)(ldsB +  row       * kStride + ko);  // B frag, N-tile 0
      v2f b1 = *(const v2f*)(ldsB + (row + 16) * kStride + ko);  // B frag, N-tile 1
      c0 = __builtin_amdgcn_wmma_f32_16x16x4_f32(false, af, false, b0,
                                                 (short)0, c0, false, false);
      c1 = __builtin_amdgcn_wmma_f32_16x16x4_f32(false, af, false, b1,
                                                 (short)0, c1, false, false);
    }

    // ---- Store D: lane(&15)=N col, VGPR i = M row (upper lane half -> M+8) ----
    if (fullTile) {
      // Unguarded stores: no exec-mask juggling per element.
      float* o = out + (rowBase + 8 * khi) * kR;
      #pragma unroll
      for (int i = 0; i < 8; ++i) {
        o[i * kR + row] = c0[i];
      }
      if (row == 0) {                       // single exec mask for all 8 stores
        #pragma unroll
        for (int i = 0; i < 8; ++i) {
          o[i * kR + 16] = c1[i];           // relation 16 (N-tile 1, col 0)
        }
      }
    } else {
      #pragma unroll
      for (int i = 0; i < 8; ++i) {
        const long long m = rowBase + i + 8 * khi;
        if (m < totalRows) {
          out[m * kR + row] = c0[i];
          if (row == 0) out[m * kR + 16] = c1[i];
        }
      }
    }
    asm volatile("" ::: "memory");  // keep this tile's LDS reads ahead of next stores
  }
}

extern "C" void kernel_launch(void* const* d_in, const int* in_sizes, int n_in,
                              void* d_out, int out_size, void* d_ws, size_t ws_size,
                              hipStream_t stream) {
  const float* z        = (const float*)d_in[0];
  const float* rel      = (const float*)d_in[1];
  const long long* posI = (const long long*)d_in[2];
  const long long* negI = (const long long*)d_in[3];
  float* out = (float*)d_out;

  const long long E = (long long)in_sizes[2] / 2;   // pos_edge_index is [2, E]
  const long long tiles = (2 * E + kTileM - 1) / kTileM;
  long long blocksNeeded = (tiles + kWaves - 1) / kWaves;
  int blocks = (int)(blocksNeeded < 2048 ? blocksNeeded : 2048);
  if (blocks < 1) blocks = 1;

  hipLaunchKernelGGL(hetdistmult_wmma, dim3(blocks), dim3(128), 0, stream,
                     z, rel, posI, negI, out, E);
}